// Stochastic_encoder_5566277616134
// MI455X (gfx1250) — compile-verified
//
#include <hip/hip_runtime.h>
#include <hip/hip_bf16.h>

typedef __attribute__((ext_vector_type(2))) float v2f;
typedef __attribute__((ext_vector_type(8))) float v8f;

// ---------------------------------------------------------------------------
// Zero fill (grid-stride)
// ---------------------------------------------------------------------------
__global__ void zero_kernel(float* __restrict__ p, size_t n) {
    size_t i = (size_t)blockIdx.x * blockDim.x + threadIdx.x;
    size_t stride = (size_t)gridDim.x * blockDim.x;
    for (; i < n; i += stride) p[i] = 0.0f;
}

// ---------------------------------------------------------------------------
// Degree count: deg[dst[e]] += 1  (hardware f32 atomic into L2)
// ---------------------------------------------------------------------------
__global__ void count_deg_kernel(const int* __restrict__ dst,
                                 float* __restrict__ deg, int E) {
    int e = blockIdx.x * blockDim.x + threadIdx.x;
    if (e < E) unsafeAtomicAdd(&deg[dst[e]], 1.0f);
}

// dinv[i] = rsqrt(deg[i] + 1)  (in place)
__global__ void make_dinv_kernel(float* __restrict__ deg, int N) {
    int i = blockIdx.x * blockDim.x + threadIdx.x;
    if (i < N) deg[i] = rsqrtf(deg[i] + 1.0f);
}

// ---------------------------------------------------------------------------
// f32 GEMM via V_WMMA_F32_16X16X4_F32.
// C[M x Ncols] = A[M x K] * B[K x Ncols], row-major, one wave per 16x16 tile.
// A frag (16x4 f32): lane m(0..15) holds A[m][k0], A[m][k0+1] with k0 = 0;
//                    lane m+16 holds k0 = 2.        (v2f per lane)
// B frag (4x16 f32): lane n(0..15) holds B[k0][n], B[k0+1][n]; lane n+16 k0=2.
// C/D (16x16 f32):   VGPR r -> row r + 8*(lane>>4), col = lane&15.
// Requires M % 16 == 0, Ncols % 16 == 0, K % 4 == 0 (true here:
// 50000 = 16*3125, 128 = 16*8, K in {256,128}).
// ---------------------------------------------------------------------------
__global__ __launch_bounds__(256)
void gemm_wmma_f32(const float* __restrict__ A, const float* __restrict__ B,
                   float* __restrict__ C, int M, int K, int Ncols) {
    const int wave = threadIdx.x >> 5;
    const int lane = threadIdx.x & 31;
    const int tilesN = Ncols >> 4;
    const int tile = blockIdx.x * (blockDim.x >> 5) + wave;
    const int tM = tile / tilesN;
    const int tN = tile - tM * tilesN;
    if (tM * 16 >= M) return;                 // wave-uniform exit

    const int mn = lane & 15;                 // A row within tile / B col within tile
    const int kp = (lane >> 4) << 1;          // 0 or 2: K sub-pair this half-wave owns

    const float* arow = A + (size_t)(tM * 16 + mn) * K;
    const float* bcol = B + (size_t)(tN * 16 + mn);

    v8f acc = {};
    for (int k = 0; k < K; k += 4) {
        v2f a, b;
        a.x = arow[k + kp];
        a.y = arow[k + kp + 1];
        b.x = bcol[(size_t)(k + kp) * Ncols];
        b.y = bcol[(size_t)(k + kp + 1) * Ncols];
        acc = __builtin_amdgcn_wmma_f32_16x16x4_f32(
            /*neg_a=*/false, a, /*neg_b=*/false, b,
            /*c_mod=*/(short)0, acc, /*reuse_a=*/false, /*reuse_b=*/false);
    }

    const int rbase = tM * 16 + ((lane >> 4) << 3);
    const int col   = tN * 16 + mn;
#pragma unroll
    for (int r = 0; r < 8; ++r)
        C[(size_t)(rbase + r) * Ncols + col] = acc[r];
}

// ---------------------------------------------------------------------------
// Edge scatter: agg[dst] += t[src] * (dinv[src]*dinv[dst]).
// One wave per edge; each lane moves a float4 (32 lanes * 4 = 128 = H).
// ---------------------------------------------------------------------------
__global__ __launch_bounds__(256)
void scatter_edges_kernel(const float* __restrict__ t, const int* __restrict__ src,
                          const int* __restrict__ dst, const float* __restrict__ dinv,
                          float* __restrict__ agg, int E) {
    const int wid  = (int)(((size_t)blockIdx.x * blockDim.x + threadIdx.x) >> 5);
    const int lane = threadIdx.x & 31;
    if (wid >= E) return;
    const int s = src[wid];
    const int d = dst[wid];
    const float w = dinv[s] * dinv[d];
    const float4 v = ((const float4*)(t + (size_t)s * 128))[lane];
    float* out = agg + (size_t)d * 128 + lane * 4;
    unsafeAtomicAdd(out + 0, v.x * w);
    unsafeAtomicAdd(out + 1, v.y * w);
    unsafeAtomicAdd(out + 2, v.z * w);
    unsafeAtomicAdd(out + 3, v.w * w);
}

// ---------------------------------------------------------------------------
// Finalize (in place on agg): agg = agg + t*dinv^2 + bias  [, ReLU]
// H = 128 features per node.
// ---------------------------------------------------------------------------
__global__ void finalize_kernel(float* __restrict__ agg, const float* __restrict__ t,
                                const float* __restrict__ dinv,
                                const float* __restrict__ bias,
                                size_t total, int relu) {
    size_t i = (size_t)blockIdx.x * blockDim.x + threadIdx.x;
    if (i >= total) return;
    const int node = (int)(i >> 7);
    const int j = (int)(i & 127);
    const float di = dinv[node];
    float v = agg[i] + t[i] * (di * di) + bias[j];
    if (relu) v = fmaxf(v, 0.0f);
    agg[i] = v;
}

// ---------------------------------------------------------------------------
// Launch
// ---------------------------------------------------------------------------
extern "C" void kernel_launch(void* const* d_in, const int* in_sizes, int n_in,
                              void* d_out, int out_size, void* d_ws, size_t ws_size,
                              hipStream_t stream) {
    const float* x   = (const float*)d_in[0];
    const int*   ei  = (const int*)d_in[1];
    const float* W1  = (const float*)d_in[2];
    const float* b1  = (const float*)d_in[3];
    const float* Wmu = (const float*)d_in[4];
    const float* bmu = (const float*)d_in[5];
    const float* Wls = (const float*)d_in[6];
    const float* bls = (const float*)d_in[7];

    const int H   = in_sizes[3];            // 128
    const int Cin = in_sizes[2] / H;        // 256
    const int N   = in_sizes[0] / Cin;      // 50000
    const int E   = in_sizes[1] / 2;        // 800000
    const int* src = ei;
    const int* dst = ei + E;

    const size_t NH = (size_t)N * H;
    float* dinv = (float*)d_ws;             // N floats
    float* t    = dinv + N;                 // N*H floats (GEMM output scratch)
    float* h    = t + NH;                   // N*H floats (layer-1 activations)
    float* mu   = (float*)d_out;            // N*H
    float* ls   = mu + NH;                  // N*H

    const int tilesPerBlock = 256 / 32;     // 8 waves per block, 1 tile per wave
    const int tiles = (N / 16) * (H / 16);
    const int gemmBlocks = (tiles + tilesPerBlock - 1) / tilesPerBlock;
    const int edgeBlocks = (E + tilesPerBlock - 1) / tilesPerBlock;  // 1 wave/edge
    const int elemBlocks = (int)((NH + 255) / 256);

    // --- degree / normalization ---
    zero_kernel<<<1024, 256, 0, stream>>>(dinv, (size_t)N);
    count_deg_kernel<<<(E + 255) / 256, 256, 0, stream>>>(dst, dinv, E);
    make_dinv_kernel<<<(N + 255) / 256, 256, 0, stream>>>(dinv, N);

    // --- layer 1: h = relu(agg(x @ W1)) ---
    gemm_wmma_f32<<<gemmBlocks, 256, 0, stream>>>(x, W1, t, N, Cin, H);
    zero_kernel<<<2048, 256, 0, stream>>>(h, NH);
    scatter_edges_kernel<<<edgeBlocks, 256, 0, stream>>>(t, src, dst, dinv, h, E);
    finalize_kernel<<<elemBlocks, 256, 0, stream>>>(h, t, dinv, b1, NH, 1);

    // --- mu head ---
    gemm_wmma_f32<<<gemmBlocks, 256, 0, stream>>>(h, Wmu, t, N, H, H);
    zero_kernel<<<2048, 256, 0, stream>>>(mu, NH);
    scatter_edges_kernel<<<edgeBlocks, 256, 0, stream>>>(t, src, dst, dinv, mu, E);
    finalize_kernel<<<elemBlocks, 256, 0, stream>>>(mu, t, dinv, bmu, NH, 0);

    // --- logstd head ---
    gemm_wmma_f32<<<gemmBlocks, 256, 0, stream>>>(h, Wls, t, N, H, H);
    zero_kernel<<<2048, 256, 0, stream>>>(ls, NH);
    scatter_edges_kernel<<<edgeBlocks, 256, 0, stream>>>(t, src, dst, dinv, ls, E);
    finalize_kernel<<<elemBlocks, 256, 0, stream>>>(ls, t, dinv, bls, NH, 0);
}